// WHTBlock_91010357002996
// MI455X (gfx1250) — compile-verified
//
#include <hip/hip_runtime.h>
#include <math.h>

// ---------------------------------------------------------------------------
// 16x16 block 2-D Walsh-Hadamard transform + threshold for MI455X (gfx1250).
// Memory-bound (805 MB traffic ~ 35us @ 23.3 TB/s); per-block math done with
// chained V_WMMA_F32_16X16X4_F32 (exact f32 16x16 shape on CDNA5).
// ---------------------------------------------------------------------------

typedef __attribute__((ext_vector_type(2))) float v2f;
typedef __attribute__((ext_vector_type(8))) float v8f;

#define IMG_W        1024
#define IMG_HW       (1024 * 1024)
#define BLOCKS_PER_IMG (64 * 64)   // 64x64 blocks of 16x16 per (b,c) image
#define NB           8             // blocks processed per wave
#define WAVES_PER_WG 8             // 256 threads / wave32
#define SCR_STRIDE   26            // LDS scratch row stride (floats): even (b64
                                   // aligned), 8*26 % 64 = 16 -> store halves in
                                   // disjoint banks, 26d==0 mod 64 => d>=32.
#define SCR_WAVE     416           // 16 * SCR_STRIDE floats per-wave scratch

__global__ __launch_bounds__(256) void wht16_wmma_kernel(
    const float* __restrict__ src, float* __restrict__ dst, int total_blocks) {
  __shared__ float scr[WAVES_PER_WG * SCR_WAVE];

  const int lane  = threadIdx.x & 31;
  const int wslot = threadIdx.x >> 5;
  const int n     = lane & 15;   // column / M index role per WMMA layouts
  const int half  = lane >> 4;   // lane-half selects K/M offsets in layouts

  float* ws = &scr[wslot * SCR_WAVE];

  // Hn registers. h[2k+t] holds Hn[n][4k + 2*half + t] (A-layout for chunk k).
  // By symmetry of Hn this is also Hn[4k + t + 2*half][n] (B-layout chunk k).
  // Hn = H/4 with H[i][j] = (-1)^popcount(i & j) (Sylvester construction).
  float h[8];
#pragma unroll
  for (int k = 0; k < 4; ++k) {
#pragma unroll
    for (int t = 0; t < 2; ++t) {
      const int K = 4 * k + 2 * half + t;
      h[2 * k + t] = (__popc(n & K) & 1) ? -0.25f : 0.25f;
    }
  }

  const long wid  = (long)blockIdx.x * WAVES_PER_WG + wslot;
  const long blk0 = wid * NB;

#pragma unroll 1
  for (int it = 0; it < NB; ++it) {
    const long blk = blk0 + it;              // wave-uniform
    if (blk >= total_blocks) break;          // wave-uniform -> EXEC stays full

    const long img = blk / BLOCKS_PER_IMG;
    const int  rem = (int)(blk - img * BLOCKS_PER_IMG);
    const int  br  = rem >> 6;               // block row in image
    const int  bc  = rem & 63;               // block col in image
    const long base = img * (long)IMG_HW + (long)(br * 16) * IMG_W + bc * 16;

    // ---- Load X in WMMA B-layout: x[2k+t] = X[4k + t + 2*half][n] ----
    const float* p = src + base + (long)(2 * half) * IMG_W + n;
    float x[8];
#pragma unroll
    for (int k = 0; k < 4; ++k) {
      x[2 * k]     = p[(4 * k) * IMG_W];
      x[2 * k + 1] = p[(4 * k + 1) * IMG_W];
    }

    // ---- Stage 1: T = Hn @ X via 4 chained 16x16x4 f32 WMMAs ----
    v8f acc = {0.f, 0.f, 0.f, 0.f, 0.f, 0.f, 0.f, 0.f};
#pragma unroll
    for (int k = 0; k < 4; ++k) {
      v2f A  = {h[2 * k], h[2 * k + 1]};
      v2f Bv = {x[2 * k], x[2 * k + 1]};
      acc = __builtin_amdgcn_wmma_f32_16x16x4_f32(
          false, A, false, Bv, (short)0, acc, false, false);
    }

    // ---- C/D-layout -> A-layout transpose through per-wave LDS scratch ----
    // Store T row-major: element (M = j + 8*half, N = n).
    {
      const int sbase = (8 * half) * SCR_STRIDE + n;
#pragma unroll
      for (int j = 0; j < 8; ++j) ws[sbase + j * SCR_STRIDE] = acc[j];
    }
    __builtin_amdgcn_wave_barrier();
    asm volatile("s_wait_dscnt 0" ::: "memory");  // DS in-order within wave;
                                                  // make cross-lane T visible.

    // A-layout chunk k: a[k] = (T[n][4k + 2*half], T[n][4k + 2*half + 1]),
    // contiguous pair -> one ds_load_b64 per chunk (8-byte aligned: stride even).
    v2f a[4];
#pragma unroll
    for (int k = 0; k < 4; ++k)
      a[k] = *(const v2f*)&ws[n * SCR_STRIDE + 4 * k + 2 * half];

    // ---- Stage 2: Y = T @ Hn (Hn symmetric -> reuse h[] as B operand) ----
    v8f y = {0.f, 0.f, 0.f, 0.f, 0.f, 0.f, 0.f, 0.f};
#pragma unroll
    for (int k = 0; k < 4; ++k) {
      v2f Bv = {h[2 * k], h[2 * k + 1]};
      y = __builtin_amdgcn_wmma_f32_16x16x4_f32(
          false, a[k], false, Bv, (short)0, y, false, false);
    }

    // ---- Threshold + store from C/D layout: element (j + 8*half, n) ----
    float* q = dst + base + (long)(8 * half) * IMG_W + n;
#pragma unroll
    for (int j = 0; j < 8; ++j) {
      float v = y[j];
      v = (fabsf(v) < 0.1f) ? 0.0f : v;
      q[j * IMG_W] = v;
    }
  }
}

extern "C" void kernel_launch(void* const* d_in, const int* in_sizes, int n_in,
                              void* d_out, int out_size, void* d_ws, size_t ws_size,
                              hipStream_t stream) {
  (void)n_in; (void)d_ws; (void)ws_size; (void)out_size;
  const float* src = (const float*)d_in[0];
  float*       dst = (float*)d_out;

  const long total_elems  = (long)in_sizes[0];      // 32*3*1024*1024
  const long total_blocks = total_elems / 256;      // 16x16 blocks: 393216
  const long total_waves  = (total_blocks + NB - 1) / NB;
  const int  wgs = (int)((total_waves + WAVES_PER_WG - 1) / WAVES_PER_WG);

  wht16_wmma_kernel<<<wgs, WAVES_PER_WG * 32, 0, stream>>>(
      src, dst, (int)total_blocks);
}